// Net_29394756173846
// MI455X (gfx1250) — compile-verified
//
#include <hip/hip_runtime.h>
#include <hip/hip_bf16.h>

#define NN   50000
#define F1   512
#define C1   64      // heads1*d1
#define H1   8
#define D1   8
#define C2   16
#define NEGS 0.2f

typedef __attribute__((ext_vector_type(2))) float v2f;
typedef __attribute__((ext_vector_type(8))) float v8f;

// ---------- monotone float <-> uint encoding for atomicMax on floats ----------
__device__ __forceinline__ unsigned fenc(float f) {
    unsigned u = __float_as_uint(f);
    return (u & 0x80000000u) ? ~u : (u | 0x80000000u);
}
__device__ __forceinline__ float fdec(unsigned u) {
    return (u & 0x80000000u) ? __uint_as_float(u & 0x7FFFFFFFu) : __uint_as_float(~u);
}

// =====================================================================
// GEMM1: H1out[N,64] = X[N,512] @ W[512,64]   via V_WMMA_F32_16X16X4_F32
// One wave per 16-row strip; 4 N-tiles of 16 -> 16x64 output per wave.
// =====================================================================
__global__ __launch_bounds__(32) void gemm1_wmma(const float* __restrict__ X,
                                                 const float* __restrict__ W,
                                                 float* __restrict__ H) {
    const int m0   = blockIdx.x * 16;
    const int lane = threadIdx.x;
    const int half = lane >> 4;          // 0: K pair (0,1), 1: K pair (2,3)
    const int l    = lane & 15;
    const int koff = half * 2;

    v8f a0 = {}, a1 = {}, a2 = {}, a3 = {};
    const float* xrow = X + (size_t)(m0 + l) * F1;

    for (int k = 0; k < F1; k += 4) {
        v2f a;
        a.x = xrow[k + koff];
        a.y = xrow[k + koff + 1];
        const float* w0 = W + (size_t)(k + koff) * C1;
        const float* w1 = w0 + C1;
        v2f b0, b1, b2, b3;
        b0.x = w0[l];       b0.y = w1[l];
        b1.x = w0[16 + l];  b1.y = w1[16 + l];
        b2.x = w0[32 + l];  b2.y = w1[32 + l];
        b3.x = w0[48 + l];  b3.y = w1[48 + l];
        a0 = __builtin_amdgcn_wmma_f32_16x16x4_f32(false, a, false, b0, (short)0, a0, false, false);
        a1 = __builtin_amdgcn_wmma_f32_16x16x4_f32(false, a, false, b1, (short)0, a1, false, false);
        a2 = __builtin_amdgcn_wmma_f32_16x16x4_f32(false, a, false, b2, (short)0, a2, false, false);
        a3 = __builtin_amdgcn_wmma_f32_16x16x4_f32(false, a, false, b3, (short)0, a3, false, false);
    }
#pragma unroll
    for (int r = 0; r < 8; ++r) {
        float* h = H + (size_t)(m0 + r + half * 8) * C1 + l;
        h[0]  = a0[r];
        h[16] = a1[r];
        h[32] = a2[r];
        h[48] = a3[r];
    }
}

// =====================================================================
// GEMM2: H2[N,16] = X2[N,64] @ W2[64,16]   via V_WMMA_F32_16X16X4_F32
// =====================================================================
__global__ __launch_bounds__(32) void gemm2_wmma(const float* __restrict__ X,
                                                 const float* __restrict__ W,
                                                 float* __restrict__ H) {
    const int m0   = blockIdx.x * 16;
    const int lane = threadIdx.x;
    const int half = lane >> 4;
    const int l    = lane & 15;
    const int koff = half * 2;

    v8f acc = {};
    const float* xrow = X + (size_t)(m0 + l) * C1;
    for (int k = 0; k < C1; k += 4) {
        v2f a;
        a.x = xrow[k + koff];
        a.y = xrow[k + koff + 1];
        v2f b;
        b.x = W[(size_t)(k + koff) * C2 + l];
        b.y = W[(size_t)(k + koff + 1) * C2 + l];
        acc = __builtin_amdgcn_wmma_f32_16x16x4_f32(false, a, false, b, (short)0, acc, false, false);
    }
#pragma unroll
    for (int r = 0; r < 8; ++r)
        H[(size_t)(m0 + r + half * 8) * C2 + l] = acc[r];
}

// ---------- per-node attention halves, layer 1: [N,8] each ----------
__global__ void alpha1_k(const float* __restrict__ h1, const float* __restrict__ asrc,
                         const float* __restrict__ adst, float* __restrict__ as1,
                         float* __restrict__ ad1) {
    int i = blockIdx.x * blockDim.x + threadIdx.x;
    if (i >= NN * H1) return;
    int node = i >> 3, h = i & 7;
    const float* hv = h1 + (size_t)node * C1 + h * D1;
    const float* av = asrc + h * D1;
    const float* bv = adst + h * D1;
    float s = 0.f, d = 0.f;
#pragma unroll
    for (int j = 0; j < D1; ++j) { s += hv[j] * av[j]; d += hv[j] * bv[j]; }
    as1[i] = s; ad1[i] = d;
}

__device__ __forceinline__ void edge_sd(const int* __restrict__ ei, int E, int e,
                                        int& s, int& d) {
    if (e < E) { s = ei[e]; d = ei[E + e]; }
    else       { s = e - E; d = e - E; }
}

// ---------- layer-1 edge passes ----------
__global__ void emax1_k(const int* __restrict__ ei, int E, const float* __restrict__ as1,
                        const float* __restrict__ ad1, unsigned* __restrict__ emax) {
    int e = blockIdx.x * blockDim.x + threadIdx.x;
    if (e >= E + NN) return;
    int s, d; edge_sd(ei, E, e, s, d);
#pragma unroll
    for (int h = 0; h < H1; ++h) {
        float v = as1[s * H1 + h] + ad1[d * H1 + h];
        v = v >= 0.f ? v : NEGS * v;
        atomicMax(emax + d * H1 + h, fenc(v));
    }
}

__global__ void esum1_k(const int* __restrict__ ei, int E, const float* __restrict__ as1,
                        const float* __restrict__ ad1, const unsigned* __restrict__ emax,
                        float* __restrict__ esum) {
    int e = blockIdx.x * blockDim.x + threadIdx.x;
    if (e >= E + NN) return;
    int s, d; edge_sd(ei, E, e, s, d);
#pragma unroll
    for (int h = 0; h < H1; ++h) {
        float v = as1[s * H1 + h] + ad1[d * H1 + h];
        v = v >= 0.f ? v : NEGS * v;
        atomicAdd(esum + d * H1 + h, expf(v - fdec(emax[d * H1 + h])));
    }
}

__global__ void agg1_k(const int* __restrict__ ei, int E, const float* __restrict__ as1,
                       const float* __restrict__ ad1, const unsigned* __restrict__ emax,
                       const float* __restrict__ esum, const float* __restrict__ h1,
                       float* __restrict__ o1) {
    int gid = blockIdx.x * blockDim.x + threadIdx.x;
    int e = gid >> 6, n = gid & 63;
    if (e >= E + NN) return;
    int s, d; edge_sd(ei, E, e, s, d);
    int h = n >> 3;
    float v = as1[s * H1 + h] + ad1[d * H1 + h];
    v = v >= 0.f ? v : NEGS * v;
    float alpha = expf(v - fdec(emax[d * H1 + h])) / esum[d * H1 + h];
    atomicAdd(o1 + (size_t)d * C1 + n, h1[(size_t)s * C1 + n] * alpha);
}

// ---------- bias + ELU (in place) ----------
__global__ void elu_k(float* __restrict__ o1, const float* __restrict__ b1) {
    int i = blockIdx.x * blockDim.x + threadIdx.x;
    if (i >= NN * C1) return;
    float v = o1[i] + b1[i & 63];
    o1[i] = v > 0.f ? v : expm1f(v);
}

// ---------- per-node attention halves, layer 2: [N] each ----------
__global__ void alpha2_k(const float* __restrict__ h2, const float* __restrict__ asrc,
                         const float* __restrict__ adst, float* __restrict__ as2,
                         float* __restrict__ ad2) {
    int i = blockIdx.x * blockDim.x + threadIdx.x;
    if (i >= NN) return;
    const float* hv = h2 + (size_t)i * C2;
    float s = 0.f, d = 0.f;
#pragma unroll
    for (int j = 0; j < C2; ++j) { s += hv[j] * asrc[j]; d += hv[j] * adst[j]; }
    as2[i] = s; ad2[i] = d;
}

// ---------- layer-2 edge passes (heads = 1) ----------
__global__ void emax2_k(const int* __restrict__ ei, int E, const float* __restrict__ as2,
                        const float* __restrict__ ad2, unsigned* __restrict__ emax) {
    int e = blockIdx.x * blockDim.x + threadIdx.x;
    if (e >= E + NN) return;
    int s, d; edge_sd(ei, E, e, s, d);
    float v = as2[s] + ad2[d];
    v = v >= 0.f ? v : NEGS * v;
    atomicMax(emax + d, fenc(v));
}

__global__ void esum2_k(const int* __restrict__ ei, int E, const float* __restrict__ as2,
                        const float* __restrict__ ad2, const unsigned* __restrict__ emax,
                        float* __restrict__ esum) {
    int e = blockIdx.x * blockDim.x + threadIdx.x;
    if (e >= E + NN) return;
    int s, d; edge_sd(ei, E, e, s, d);
    float v = as2[s] + ad2[d];
    v = v >= 0.f ? v : NEGS * v;
    atomicAdd(esum + d, expf(v - fdec(emax[d])));
}

__global__ void agg2_k(const int* __restrict__ ei, int E, const float* __restrict__ as2,
                       const float* __restrict__ ad2, const unsigned* __restrict__ emax,
                       const float* __restrict__ esum, const float* __restrict__ h2,
                       float* __restrict__ out) {
    int gid = blockIdx.x * blockDim.x + threadIdx.x;
    int e = gid >> 4, n = gid & 15;
    if (e >= E + NN) return;
    int s, d; edge_sd(ei, E, e, s, d);
    float v = as2[s] + ad2[d];
    v = v >= 0.f ? v : NEGS * v;
    float alpha = expf(v - fdec(emax[d])) / esum[d];
    atomicAdd(out + (size_t)d * C2 + n, h2[(size_t)s * C2 + n] * alpha);
}

// ---------- bias + log_softmax over each 16-wide row, in place ----------
__global__ void lsm_k(float* __restrict__ out, const float* __restrict__ b2) {
    int i = blockIdx.x * blockDim.x + threadIdx.x;
    if (i >= NN) return;
    float* row = out + (size_t)i * C2;
    float v[C2];
    float m = -3.402823466e38f;
#pragma unroll
    for (int j = 0; j < C2; ++j) { v[j] = row[j] + b2[j]; m = fmaxf(m, v[j]); }
    float s = 0.f;
#pragma unroll
    for (int j = 0; j < C2; ++j) s += expf(v[j] - m);
    float ls = logf(s);
#pragma unroll
    for (int j = 0; j < C2; ++j) row[j] = v[j] - m - ls;
}

extern "C" void kernel_launch(void* const* d_in, const int* in_sizes, int n_in,
                              void* d_out, int out_size, void* d_ws, size_t ws_size,
                              hipStream_t stream) {
    const float* x     = (const float*)d_in[0];
    const int*   ei    = (const int*)  d_in[1];
    const float* W1    = (const float*)d_in[2];
    const float* asrc1 = (const float*)d_in[3];
    const float* adst1 = (const float*)d_in[4];
    const float* b1    = (const float*)d_in[5];
    const float* W2    = (const float*)d_in[6];
    const float* asrc2 = (const float*)d_in[7];
    const float* adst2 = (const float*)d_in[8];
    const float* b2    = (const float*)d_in[9];
    const int E  = in_sizes[1] / 2;      // 1,200,000
    const int ET = E + NN;               // + self-loops

    // workspace carve-up (~36 MB)
    float*    h1    = (float*)d_ws;             // N*64
    float*    as1   = h1  + (size_t)NN * C1;    // N*8
    float*    ad1   = as1 + (size_t)NN * H1;    // N*8
    unsigned* emax1 = (unsigned*)(ad1 + (size_t)NN * H1);  // N*8
    float*    esum1 = (float*)(emax1 + (size_t)NN * H1);   // N*8
    float*    o1    = esum1 + (size_t)NN * H1;  // N*64
    float*    h2    = o1  + (size_t)NN * C1;    // N*16
    float*    as2   = h2  + (size_t)NN * C2;    // N
    float*    ad2   = as2 + NN;                 // N
    unsigned* emax2 = (unsigned*)(ad2 + NN);    // N
    float*    esum2 = (float*)(emax2 + NN);     // N

    // per-call reset of accumulation targets (graph-capture safe)
    hipMemsetAsync(emax1, 0, (size_t)NN * H1 * 4, stream);
    hipMemsetAsync(esum1, 0, (size_t)NN * H1 * 4, stream);
    hipMemsetAsync(o1,    0, (size_t)NN * C1 * 4, stream);
    hipMemsetAsync(emax2, 0, (size_t)NN * 4, stream);
    hipMemsetAsync(esum2, 0, (size_t)NN * 4, stream);
    hipMemsetAsync(d_out, 0, (size_t)out_size * 4, stream);

    const int TB = 256;
    // layer 1
    gemm1_wmma<<<NN / 16, 32, 0, stream>>>(x, W1, h1);
    alpha1_k<<<(NN * H1 + TB - 1) / TB, TB, 0, stream>>>(h1, asrc1, adst1, as1, ad1);
    emax1_k<<<(ET + TB - 1) / TB, TB, 0, stream>>>(ei, E, as1, ad1, emax1);
    esum1_k<<<(ET + TB - 1) / TB, TB, 0, stream>>>(ei, E, as1, ad1, emax1, esum1);
    agg1_k<<<((long long)ET * C1 + TB - 1) / TB, TB, 0, stream>>>(ei, E, as1, ad1, emax1, esum1, h1, o1);
    elu_k<<<(NN * C1 + TB - 1) / TB, TB, 0, stream>>>(o1, b1);
    // layer 2
    gemm2_wmma<<<NN / 16, 32, 0, stream>>>(o1, W2, h2);
    alpha2_k<<<(NN + TB - 1) / TB, TB, 0, stream>>>(h2, asrc2, adst2, as2, ad2);
    emax2_k<<<(ET + TB - 1) / TB, TB, 0, stream>>>(ei, E, as2, ad2, emax2);
    esum2_k<<<(ET + TB - 1) / TB, TB, 0, stream>>>(ei, E, as2, ad2, emax2, esum2);
    agg2_k<<<((long long)ET * C2 + TB - 1) / TB, TB, 0, stream>>>(ei, E, as2, ad2, emax2, esum2, h2, (float*)d_out);
    lsm_k<<<(NN + TB - 1) / TB, TB, 0, stream>>>((float*)d_out, b2);
}